// MPNN_Multidim_67302137528968
// MI455X (gfx1250) — compile-verified
//
#include <hip/hip_runtime.h>
#include <hip/hip_bf16.h>

#define N_OBS 7
#define DIMW 135
#define NN 128
#define FF 64
#define NL 3
#define OD 4
#define BATCH 32

typedef __attribute__((ext_vector_type(2))) float v2f;
typedef __attribute__((ext_vector_type(8))) float v8f;

// Pre-pass: norm[b][j] = clamp(count_i(adj[b][i][j] != 0), >=1)
__global__ __launch_bounds__(128) void mpnn_norm_kernel(const float* __restrict__ obs,
                                                        float* __restrict__ norms) {
  const int b = blockIdx.x;
  const int j = threadIdx.x;
  const float* ob = obs + (size_t)b * NN * DIMW;
  int cnt = 0;
  for (int i = 0; i < NN; ++i)
    cnt += (ob[i * DIMW + N_OBS + j] != 0.0f) ? 1 : 0;
  norms[b * NN + j] = (cnt == 0) ? 1.0f : (float)cnt;
}

// Wave-level f32 GEMM: one wave owns a 16-row M-tile, sweeps 4 N-tiles (N=64),
// K-loop of 4 per V_WMMA_F32_16X16X4_F32.
// A layout: lane = (k>=2?16:0)+m, vgpr = k&1  -> a.{x,y} = A[m][k0], A[m][k0+1], k0 = 2*(lane>>4)
// B layout (symmetric): b.{x,y} = B[k0][n], B[k0+1][n], n = lane&15
// C/D: vgpr r, M = r + 8*(lane>>4), N = lane&15
template <class FA, class FB, class FS>
__device__ __forceinline__ void wave_gemm_m16(int K, int lane, int m0, FA la, FB lb, FS st) {
  const int lm = lane & 15;
  const int kh = (lane >> 4) << 1;
  const int mb = m0 + ((lane >> 4) << 3);
  for (int nt = 0; nt < 4; ++nt) {
    const int n = nt * 16 + lm;
    v8f c = {};
    for (int kk = 0; kk < K; kk += 4) {
      const int k0 = kk + kh;
      v2f a, bv;
      a.x = la(m0 + lm, k0);
      a.y = la(m0 + lm, k0 + 1);
      bv.x = lb(k0, n);
      bv.y = lb(k0 + 1, n);
      c = __builtin_amdgcn_wmma_f32_16x16x4_f32(false, a, false, bv, (short)0, c,
                                                false, false);
    }
#pragma unroll
    for (int r = 0; r < 8; ++r) st(mb + r, n, c[r]);
  }
}

__global__ __launch_bounds__(256) void mpnn_main_kernel(
    const float* __restrict__ obs, const float* __restrict__ W_init,
    const float* __restrict__ W_ee, const float* __restrict__ W_ef,
    const float* __restrict__ W_msg, const float* __restrict__ W_upd,
    const float* __restrict__ W_pool, const float* __restrict__ W_read,
    const float* __restrict__ b_read, const float* __restrict__ norms,
    float* __restrict__ out) {
  __shared__ float sCur[NN][FF];   // node features (ping)
  __shared__ float sEmb[NN][FF];   // edge_emb (persistent)
  __shared__ float sAgg[NN][FF];   // agg / next-cur (pong)
  __shared__ float sMsg[NN][FF];   // msg; reused as P (edge proj) early
  __shared__ float sNF[NN][8];     // raw node features (7 used)
  __shared__ float sNorm[NN];
  __shared__ float sRed[256];
  __shared__ float sPool[FF];
  __shared__ float sHP[FF];
  __shared__ float sAo[NN][OD];
  __shared__ float sBo[NN][OD];
  __shared__ float sC0[OD];

  const int tid = threadIdx.x;
  const int lane = tid & 31;
  const int wv = tid >> 5;
  const int b = blockIdx.x;
  const float* ob = obs + (size_t)b * NN * DIMW;

  // ---- Phase A: stage node features, norms; global norm.max() ----
  for (int idx = tid; idx < NN * N_OBS; idx += 256) {
    const int i = idx / N_OBS, k = idx % N_OBS;
    sNF[i][k] = ob[i * DIMW + k];
  }
  if (tid < NN) sNorm[tid] = norms[b * NN + tid];
  float mx = 1.0f;
  for (int idx = tid; idx < BATCH * NN; idx += 256) mx = fmaxf(mx, norms[idx]);
  sRed[tid] = mx;
  __syncthreads();
  for (int s = 128; s > 0; s >>= 1) {
    if (tid < s) sRed[tid] = fmaxf(sRed[tid], sRed[tid + s]);
    __syncthreads();
  }
  const float invmax = 1.0f / sRed[0];

  // ---- Phase B: init_emb = relu(nf @ W_init^T) -> sCur ----
  for (int idx = tid; idx < NN * FF; idx += 256) {
    const int i = idx >> 6, f = idx & 63;
    float acc = 0.0f;
#pragma unroll
    for (int k = 0; k < N_OBS; ++k) acc += sNF[i][k] * W_init[f * N_OBS + k];
    sCur[i][f] = fmaxf(acc, 0.0f);
  }

  // ---- Phase C: P[j][f] = nf[j] . W_ee[f][1:8] -> sMsg ----
  for (int idx = tid; idx < NN * FF; idx += 256) {
    const int j = idx >> 6, f = idx & 63;
    float acc = 0.0f;
    if (f < 63) {
#pragma unroll
      for (int k = 0; k < N_OBS; ++k) acc += sNF[j][k] * W_ee[f * 8 + 1 + k];
    }
    sMsg[j][f] = acc;
  }
  __syncthreads();

  // ---- Phase D: S[i][f] = sum_j[adj!=0] relu(adj*w0 + P[j][f]) / norm[i] -> sAgg ----
  {
    const int f = tid & 63;
    const int ib = tid >> 6;  // uniform per wave -> adj load is a broadcast
    const float w0 = (f < 63) ? W_ee[f * 8] : 0.0f;
    for (int i = ib; i < NN; i += 4) {
      float acc = 0.0f;
      if (f < 63) {
        for (int j = 0; j < NN; ++j) {
          const float a = ob[i * DIMW + N_OBS + j];
          if (a != 0.0f) acc += fmaxf(a * w0 + sMsg[j][f], 0.0f);
        }
      }
      sAgg[i][f] = acc / sNorm[i];
    }
  }
  __syncthreads();

  // ---- Phase E: edge_emb = relu(V @ W_ef^T), V = [S | norm/normmax] (WMMA) ----
  wave_gemm_m16(
      FF, lane, wv * 16,
      [&](int m, int k) { return (k < 63) ? sAgg[m][k] : sNorm[m] * invmax; },
      [&](int k, int n) { return W_ef[n * FF + k]; },
      [&](int m, int n, float v) { sEmb[m][n] = fmaxf(v, 0.0f); });
  __syncthreads();

  // ---- Message-passing layers (all GEMMs via WMMA f32 16x16x4) ----
  float(*cur)[FF] = sCur;
  float(*tmp)[FF] = sAgg;
  for (int L = 0; L < NL; ++L) {
    const float* Wm = W_msg + (size_t)L * FF * 2 * FF;
    const float* Wu = W_upd + (size_t)L * FF * 2 * FF;
    __builtin_prefetch(Wm, 0, 1);
    __builtin_prefetch(Wu, 0, 1);
    // agg = adj @ cur / norm  (A from global adj, B from LDS cur)
    wave_gemm_m16(
        NN, lane, wv * 16,
        [&](int m, int k) { return ob[m * DIMW + N_OBS + k]; },
        [&](int k, int n) { return cur[k][n]; },
        [&](int m, int n, float v) { tmp[m][n] = v / sNorm[m]; });
    __syncthreads();
    // msg = relu([agg | edge_emb] @ Wm^T)
    wave_gemm_m16(
        2 * FF, lane, wv * 16,
        [&](int m, int k) { return (k < FF) ? tmp[m][k] : sEmb[m][k - FF]; },
        [&](int k, int n) { return Wm[n * 2 * FF + k]; },
        [&](int m, int n, float v) { sMsg[m][n] = fmaxf(v, 0.0f); });
    __syncthreads();
    // cur' = relu([cur | msg] @ Wu^T)  (overwrites dead agg buffer)
    wave_gemm_m16(
        2 * FF, lane, wv * 16,
        [&](int m, int k) { return (k < FF) ? cur[m][k] : sMsg[m][k - FF]; },
        [&](int k, int n) { return Wu[n * 2 * FF + k]; },
        [&](int m, int n, float v) { tmp[m][n] = fmaxf(v, 0.0f); });
    __syncthreads();
    float(*t)[FF] = cur;
    cur = tmp;
    tmp = t;
  }

  // ---- Readout: out[i,j,:] = c0 + A[i] + B[j] ----
  if (tid < FF) {
    float s = 0.0f;
    for (int i = 0; i < NN; ++i) s += cur[i][tid];
    sPool[tid] = s * (1.0f / NN);
  }
  __syncthreads();
  if (tid < FF) {
    float s = 0.0f;
    for (int f = 0; f < FF; ++f) s += sPool[f] * W_pool[tid * FF + f];
    sHP[tid] = fmaxf(s, 0.0f);  // relu(h_pooled)
  }
  __syncthreads();
  if (tid < OD) {
    float s = b_read[tid];
    for (int f = 0; f < FF; ++f) s += W_read[tid * 3 * FF + f] * sHP[f];
    sC0[tid] = s;
  }
  for (int idx = tid; idx < NN * OD; idx += 256) {
    const int i = idx >> 2, o = idx & 3;
    const float* wr = W_read + o * 3 * FF;
    float sa = 0.0f, sb = 0.0f;
    for (int f = 0; f < FF; ++f) {
      const float cf = cur[i][f];
      sa += wr[FF + f] * cf;
      sb += wr[2 * FF + f] * cf;
    }
    sAo[i][o] = sa;
    sBo[i][o] = sb;
  }
  __syncthreads();
  float4* out4 = (float4*)(out + (size_t)b * NN * NN * OD);
  const float c00 = sC0[0], c01 = sC0[1], c02 = sC0[2], c03 = sC0[3];
  for (int idx = tid; idx < NN * NN; idx += 256) {
    const int i = idx >> 7, j = idx & 127;
    float4 r;
    r.x = c00 + sAo[i][0] + sBo[j][0];
    r.y = c01 + sAo[i][1] + sBo[j][1];
    r.z = c02 + sAo[i][2] + sBo[j][2];
    r.w = c03 + sAo[i][3] + sBo[j][3];
    out4[idx] = r;
  }
}

extern "C" void kernel_launch(void* const* d_in, const int* in_sizes, int n_in,
                              void* d_out, int out_size, void* d_ws, size_t ws_size,
                              hipStream_t stream) {
  const float* obs    = (const float*)d_in[0];
  const float* W_init = (const float*)d_in[1];
  const float* W_ee   = (const float*)d_in[2];
  const float* W_ef   = (const float*)d_in[3];
  const float* W_msg  = (const float*)d_in[4];
  const float* W_upd  = (const float*)d_in[5];
  const float* W_pool = (const float*)d_in[6];
  const float* W_read = (const float*)d_in[7];
  const float* b_read = (const float*)d_in[8];
  float* out = (float*)d_out;
  float* norms = (float*)d_ws;  // BATCH*NN floats = 16 KB

  mpnn_norm_kernel<<<BATCH, NN, 0, stream>>>(obs, norms);
  mpnn_main_kernel<<<BATCH, 256, 0, stream>>>(obs, W_init, W_ee, W_ef, W_msg,
                                              W_upd, W_pool, W_read, b_read,
                                              norms, out);
}